// Layer_29351806501586
// MI455X (gfx1250) — compile-verified
//
#include <hip/hip_runtime.h>
#include <stdint.h>

// Problem constants from the reference file.
#define BB 16384   // batch
#define DD 8192    // wires (data dim)
#define GG 16384   // gates
#define BT (BB / 32)  // 512 batch bit-plane tiles

#define THREADS 256
#define GPV 4                                   // consecutive gates per thread-vector
#define KITER 2                                 // vectors per thread
#define GATES_PER_BLOCK (THREADS * GPV * KITER) // 2048

typedef uint32_t v4u __attribute__((ext_vector_type(4)));
typedef uint32_t v8u __attribute__((ext_vector_type(8)));
typedef float    v4f __attribute__((ext_vector_type(4)));

// ---------------------------------------------------------------------------
// Kernel 1: bit-pack input_values [B, D] (bool bytes, 0/1) into batch
// bit-planes packed[B/32, D] : bit j of packed[bt, d] = input_values[bt*32+j, d].
// Each thread handles 4 consecutive wire columns: 32 coalesced b32 loads
// (non-temporal: input is read exactly once) -> one uint4 (b128) store
// (regular temporal: packed planes are reused 8x and by the TDM).
// ---------------------------------------------------------------------------
__global__ void __launch_bounds__(THREADS) pack_kernel(
    const uint8_t* __restrict__ in, uint32_t* __restrict__ packed)
{
    const int d0 = (blockIdx.x * THREADS + threadIdx.x) * 4;
    const int bt = blockIdx.y;
    const uint8_t* base = in + (size_t)bt * 32u * DD + d0;
    uint32_t w0 = 0u, w1 = 0u, w2 = 0u, w3 = 0u;
#pragma unroll
    for (int j = 0; j < 32; ++j) {
        const uint32_t q =
            __builtin_nontemporal_load((const uint32_t*)(base + (size_t)j * DD));
        w0 |= ( q         & 1u) << j;
        w1 |= ((q >>  8u) & 1u) << j;
        w2 |= ((q >> 16u) & 1u) << j;
        w3 |= ((q >> 24u) & 1u) << j;
    }
    v4u w;
    w.x = w0; w.y = w1; w.z = w2; w.w = w3;
    *(v4u*)(packed + (size_t)bt * DD + d0) = w;
}

// ---------------------------------------------------------------------------
// Kernel 2: evaluate all gates for one 32-sample batch tile per block.
//  - Wave 0 issues ONE tensor_load_to_lds DMA of the 32 KB packed wire tile
//    (TDM, TENSORcnt-tracked), waits s_wait_tensorcnt 0, then barrier.
//  - Per gate: two ds_load_b32 gathers + 4-term bitwise LUT over 32 samples.
//  - Expand bits to floats; non-temporal float4 stores (512 B/wave, write-once
//    stream that must not evict the L2-resident packed planes / descriptors).
// ---------------------------------------------------------------------------
__global__ void __launch_bounds__(THREADS) eval_kernel(
    const uint32_t* __restrict__ packed,
    const int*      __restrict__ idxs,   // [G, 2]
    const int*      __restrict__ ops,    // [G]
    float*          __restrict__ out)    // [B, G]
{
    __shared__ uint32_t wires[DD];  // 32 KB per block (320 KB/WGP available)

    const int bt    = blockIdx.y;
    const int gbase = blockIdx.x * GATES_PER_BLOCK;
    const int tid   = threadIdx.x;

    // ---- TDM: stage packed[bt, 0:D] -> LDS (one DMA for the whole tile) ----
    if (tid < 32) {  // wave 0 only; TDM ignores EXEC, issues once per wave
        const uint32_t lds_addr = (uint32_t)(uintptr_t)(&wires[0]);
        const uint64_t ga = (uint64_t)(uintptr_t)(packed + (size_t)bt * DD);

        // D# group 0 (4 SGPRs): count=1 | lds_addr | global_addr[56:0] | type=2
        v4u g0;
        g0.x = 1u;                                    // count=1, user mode
        g0.y = lds_addr;                              // LDS byte address
        g0.z = (uint32_t)ga;                          // global_addr[31:0]
        g0.w = ((uint32_t)(ga >> 32) & 0x01FFFFFFu)   // global_addr[56:32]
               | 0x80000000u;                         // type=2 (bits 127:126 = 10b)

        // D# group 1 (8 SGPRs): wg_mask=0, data_size=4B, dims for a 1-row
        // 2D tile of 8192 dwords (32 KB), stride 8192 dwords.
        v8u g1;
        g1[0] = 0x00020000u;          // data_size=2 (4B) in bits 17:16
        g1[1] = (uint32_t)DD << 16;   // tensor_dim0[15:0]=8192 in bits 63:48
        g1[2] = 0x00010000u;          // tensor_dim1=1 in bits 95:80
        g1[3] = (uint32_t)DD << 16;   // tile_dim0=8192 in bits 127:112
        g1[4] = 1u;                   // tile_dim1=1 in bits 143:128
        g1[5] = (uint32_t)DD;         // tensor_dim0_stride[31:0]=8192
        g1[6] = 0u;                   // stride hi / tensor_dim1_stride lo
        g1[7] = 0u;                   // tensor_dim1_stride hi

        asm volatile("tensor_load_to_lds %0, %1" :: "s"(g0), "s"(g1) : "memory");
        __builtin_amdgcn_s_wait_tensorcnt(0);
    }
    __syncthreads();

    // ---- Evaluate gates ----
#pragma unroll
    for (int k = 0; k < KITER; ++k) {
        const int g = gbase + (k * THREADS + tid) * GPV;  // 4 consecutive gates

        // Prefetch next iteration's gate descriptors into L2 (global_prefetch_b8).
        __builtin_prefetch(idxs + 2 * (size_t)(g + THREADS * GPV), 0, 0);

        // Coalesced descriptor loads: 8 idx ints (2x b128) + 4 op ints (b128).
        const v4u q0 = ((const v4u*)(idxs + 2 * (size_t)g))[0];
        const v4u q1 = ((const v4u*)(idxs + 2 * (size_t)g))[1];
        const v4u o4 = *(const v4u*)(ops + (size_t)g);

        const uint32_t ia[GPV] = {q0.x, q0.z, q1.x, q1.z};
        const uint32_t ib[GPV] = {q0.y, q0.w, q1.y, q1.w};
        const uint32_t op[GPV] = {o4.x, o4.y, o4.z, o4.w};

        uint32_t w[GPV];
#pragma unroll
        for (int u = 0; u < GPV; ++u) {
            const uint32_t a = wires[ia[u]];   // ds_load_b32 gather
            const uint32_t b = wires[ib[u]];   // ds_load_b32 gather
            const uint32_t o = op[u];
            // LUT bit k = output for code k = a*2+b  (matches reference LUTS).
            const uint32_t m00 = 0u - (o         & 1u);
            const uint32_t m01 = 0u - ((o >> 1u) & 1u);
            const uint32_t m10 = 0u - ((o >> 2u) & 1u);
            const uint32_t m11 = 0u - ((o >> 3u) & 1u);
            w[u] = (m00 & ~a & ~b) | (m01 & ~a & b) |
                   (m10 &  a & ~b) | (m11 &  a & b);
        }

        // Expand 32 batch bits -> float rows. For fixed j the wave stores
        // 32 lanes x float4 = 512 contiguous bytes, non-temporal.
        float* orow = out + ((size_t)bt * 32u) * GG + (size_t)g;
#pragma unroll 4
        for (int j = 0; j < 32; ++j) {
            v4f v;
            v.x = (w[0] >> j) & 1u ? 1.0f : 0.0f;
            v.y = (w[1] >> j) & 1u ? 1.0f : 0.0f;
            v.z = (w[2] >> j) & 1u ? 1.0f : 0.0f;
            v.w = (w[3] >> j) & 1u ? 1.0f : 0.0f;
            __builtin_nontemporal_store(v, (v4f*)(orow + (size_t)j * GG));
        }
    }
}

// ---------------------------------------------------------------------------
// Host-side launcher.
//   d_in[0]: input_values  [B, D]  bool (1 byte each)
//   d_in[1]: input_idxs    [G, 2]  int32
//   d_in[2]: op_ids        [G]     int32
//   d_out  : [B, G] float
//   d_ws   : >= (B/32)*D*4 = 16 MB used for packed bit-planes
// ---------------------------------------------------------------------------
extern "C" void kernel_launch(void* const* d_in, const int* in_sizes, int n_in,
                              void* d_out, int out_size, void* d_ws, size_t ws_size,
                              hipStream_t stream)
{
    const uint8_t* input_values = (const uint8_t*)d_in[0];
    const int*     input_idxs   = (const int*)d_in[1];
    const int*     op_ids       = (const int*)d_in[2];
    float*         out          = (float*)d_out;
    uint32_t*      packed       = (uint32_t*)d_ws;

    dim3 pgrid(DD / (4 * THREADS), BT);   // 8 x 512
    pack_kernel<<<pgrid, THREADS, 0, stream>>>(input_values, packed);

    dim3 egrid(GG / GATES_PER_BLOCK, BT); // 8 x 512
    eval_kernel<<<egrid, THREADS, 0, stream>>>(packed, input_idxs, op_ids, out);
}